// SDGraphUNet_1821066133676
// MI455X (gfx1250) — compile-verified
//
#include <hip/hip_runtime.h>
#include <hip/hip_bf16.h>

typedef __attribute__((ext_vector_type(2))) float v2f;
typedef __attribute__((ext_vector_type(8))) float v8f;

#define N_STK 1024
#define N_PNT 32
#define SP_C 128
#define DN_C 128
#define COOR 32
#define M_CTR 512
#define BS 8
#define XP_STRIDE 48   // float2 stride: 2*48 mod 64 == 32 -> lane halves on disjoint banks
#define ZP_STRIDE 16   // float2 stride: 2*16 mod 64 == 32 -> same property

__device__ __forceinline__ v8f wmma4(v2f a, v2f b, v8f c) {
    // D = A(16x4 f32) * B(4x16 f32) + C(16x16 f32), full f32 precision
    return __builtin_amdgcn_wmma_f32_16x16x4_f32(false, a, false, b, (short)0, c,
                                                 false, false);
}

// ---------------------------------------------------------------------------
// Kernel 1: farthest point sampling. One block (1024 threads) per batch.
// Point cloud lives in LDS; per-step argmax via wave32 shuffle + LDS tree.
// Also emits stk_coor_sampled (output 3).
// ---------------------------------------------------------------------------
extern "C" __global__ __launch_bounds__(1024)
void fps_kernel(const float* __restrict__ coor, int* __restrict__ fpsIdx,
                float* __restrict__ outCoor) {
    extern __shared__ float smem[];
    float* cSh   = smem;                          // [1024][33]
    int*   idxSh = (int*)(smem + N_STK * 33);     // [512]
    float* wv    = (float*)(idxSh + M_CTR);       // [32]
    int*   wi    = (int*)(wv + 32);               // [32]
    int*   lastSh = wi + 32;                      // [1]

    const int b = blockIdx.x;
    const int t = threadIdx.x;
    const int lane = t & 31, wid = t >> 5;
    const float* cb = coor + (size_t)b * N_STK * COOR;

    for (int e = t; e < N_STK * COOR; e += 1024)
        cSh[(e >> 5) * 33 + (e & 31)] = cb[e];
    if (t == 0) idxSh[0] = 0;
    __syncthreads();

    float dmin = 1e10f;
    int last = 0;
    for (int s = 1; s < M_CTR; ++s) {
        float d2 = 0.f;
        const float* p = cSh + last * 33;
        const float* q = cSh + t * 33;
#pragma unroll
        for (int d = 0; d < COOR; ++d) { float df = q[d] - p[d]; d2 += df * df; }
        dmin = fminf(dmin, d2);

        float v = dmin; int i = t;
#pragma unroll
        for (int mask = 16; mask >= 1; mask >>= 1) {
            float ov = __shfl_xor(v, mask, 32);
            int   oi = __shfl_xor(i, mask, 32);
            if (ov > v || (ov == v && oi < i)) { v = ov; i = oi; }
        }
        if (lane == 0) { wv[wid] = v; wi[wid] = i; }
        __syncthreads();
        if (wid == 0) {
            v = wv[lane]; i = wi[lane];
#pragma unroll
            for (int mask = 16; mask >= 1; mask >>= 1) {
                float ov = __shfl_xor(v, mask, 32);
                int   oi = __shfl_xor(i, mask, 32);
                if (ov > v || (ov == v && oi < i)) { v = ov; i = oi; }
            }
            if (lane == 0) { lastSh[0] = i; idxSh[s] = i; }
        }
        __syncthreads();
        last = lastSh[0];
    }

    if (t < M_CTR) fpsIdx[b * M_CTR + t] = idxSh[t];
    for (int e = t; e < M_CTR * COOR; e += 1024) {
        int m = e >> 5, d = e & 31;
        outCoor[(size_t)b * M_CTR * COOR + e] = cSh[idxSh[m] * 33 + d];
    }
}

// ---------------------------------------------------------------------------
// Kernel 2: 2-NN over all 1024 strokes, only for the 512 FPS centers.
// One wave per center, 8 waves per block; coords + x2 staged in LDS.
// ---------------------------------------------------------------------------
extern "C" __global__ __launch_bounds__(256)
void knn_kernel(const float* __restrict__ coor, const int* __restrict__ fpsIdx,
                int* __restrict__ knnIdx) {
    extern __shared__ float smem[];
    float* cSh = smem;                 // [1024][33]
    float* x2  = smem + N_STK * 33;    // [1024]

    const int b = blockIdx.x >> 6;
    const int mBase = (blockIdx.x & 63) * 8;
    const int t = threadIdx.x, lane = t & 31, wid = t >> 5;
    const float* cb = coor + (size_t)b * N_STK * COOR;

    for (int e = t; e < N_STK * COOR; e += 256)
        cSh[(e >> 5) * 33 + (e & 31)] = cb[e];
    __syncthreads();
    for (int p = t; p < N_STK; p += 256) {
        float s = 0.f;
        for (int d = 0; d < COOR; ++d) { float c = cSh[p * 33 + d]; s += c * c; }
        x2[p] = s;
    }
    __syncthreads();

    const int m   = mBase + wid;
    const int ctr = fpsIdx[b * M_CTR + m];
    const float x2c = x2[ctr];
    float d0 = 3.4e38f, d1 = 3.4e38f;
    int   i0 = 0x7fffffff, i1 = 0x7fffffff;
    for (int p = lane; p < N_STK; p += 32) {
        float dot = 0.f;
#pragma unroll 8
        for (int d = 0; d < COOR; ++d) dot += cSh[p * 33 + d] * cSh[ctr * 33 + d];
        float dd = x2c + x2[p] - 2.f * dot;
        if (dd < d0 || (dd == d0 && p < i0)) { d1 = d0; i1 = i0; d0 = dd; i0 = p; }
        else if (dd < d1 || (dd == d1 && p < i1)) { d1 = dd; i1 = p; }
    }
#pragma unroll
    for (int mask = 16; mask >= 1; mask >>= 1) {
        float e0 = __shfl_xor(d0, mask, 32); int j0 = __shfl_xor(i0, mask, 32);
        float e1 = __shfl_xor(d1, mask, 32); int j1 = __shfl_xor(i1, mask, 32);
        float n0, n1; int k0, k1;
        bool mineFirst = (d0 < e0) || (d0 == e0 && i0 < j0);
        if (mineFirst) {
            n0 = d0; k0 = i0;
            bool t2 = (d1 < e0) || (d1 == e0 && i1 < j0);
            if (t2) { n1 = d1; k1 = i1; } else { n1 = e0; k1 = j0; }
        } else {
            n0 = e0; k0 = j0;
            bool t2 = (d0 < e1) || (d0 == e1 && i0 < j1);
            if (t2) { n1 = d0; k1 = i0; } else { n1 = e1; k1 = j1; }
        }
        d0 = n0; i0 = k0; d1 = n1; i1 = k1;
    }
    if (lane == 0) {
        knnIdx[((size_t)b * M_CTR + m) * 2 + 0] = i0;
        knnIdx[((size_t)b * M_CTR + m) * 2 + 1] = i1;
    }
}

// ---------------------------------------------------------------------------
// Kernel 3: sparse branch. Gather cat into a pair-packed LDS panel
// (Xp[c/2][col] float2), WMMA GEMM with b64 LDS B-frags and b64 global
// A-frags, BN+ReLU epilogue, max over K via shfl_xor(1).
// One block per (batch, group of 32 centers).
// ---------------------------------------------------------------------------
extern "C" __global__ __launch_bounds__(256)
void sparse_kernel(const float* __restrict__ sf, const int* __restrict__ fpsIdx,
                   const int* __restrict__ knnIdx,
                   const float* __restrict__ Wsp, const float* __restrict__ bsp,
                   const float* __restrict__ ssp, const float* __restrict__ hsp,
                   float* __restrict__ out0) {
    extern __shared__ float smem[];
    float* X    = smem;                               // [128 pair-rows][XP_STRIDE f2]
    int*   ctrL = (int*)(X + 128 * XP_STRIDE * 2);    // [32]
    int*   nbL  = ctrL + 32;                          // [64]

    const int b  = blockIdx.x >> 4;
    const int m0 = (blockIdx.x & 15) * 32;
    const int t = threadIdx.x, lane = t & 31, wid = t >> 5;
    const int half = lane >> 4, row = lane & 15;

    if (t < 32) ctrL[t] = fpsIdx[b * M_CTR + m0 + t];
    if (t < 64) nbL[t]  = knnIdx[((size_t)b * M_CTR + m0 + (t >> 1)) * 2 + (t & 1)];
    __syncthreads();

    const float* sb = sf + (size_t)b * SP_C * N_STK;
    for (int e = t; e < 256 * 64; e += 256) {
        int c = e >> 6, col = e & 63;
        int ctr = ctrL[col >> 1];
        float v;
        if (c < 128) v = sb[c * N_STK + nbL[col]] - sb[c * N_STK + ctr];
        else         v = sb[(c - 128) * N_STK + ctr];
        X[((c >> 1) * XP_STRIDE + col) * 2 + (c & 1)] = v;
    }
    __syncthreads();

    const int o0 = wid * 16;
    v8f acc[4] = {};
    const v2f* Wrow = (const v2f*)(Wsp + (o0 + row) * 256);
    const v2f* Xp   = (const v2f*)X;
    for (int kk = 0; kk < 64; ++kk) {
        int ch = kk * 2 + half;              // c0/2, c0 = 4*kk + 2*half
        v2f a = Wrow[ch];
#pragma unroll
        for (int ct = 0; ct < 4; ++ct) {
            v2f bf = Xp[ch * XP_STRIDE + ct * 16 + row];
            acc[ct] = wmma4(a, bf, acc[ct]);
        }
    }
#pragma unroll
    for (int ct = 0; ct < 4; ++ct) {
#pragma unroll
        for (int v = 0; v < 8; ++v) {
            int o = o0 + v + 8 * half;
            int colg = ct * 16 + row;
            float tv = acc[ct][v] + bsp[o];
            tv = fmaxf(tv * ssp[o] + hsp[o], 0.f);
            float ov = __shfl_xor(tv, 1, 32);
            float r  = fmaxf(tv, ov);
            if ((lane & 1) == 0) {
                int mg = m0 + (colg >> 1);
                out0[(size_t)b * SP_C * M_CTR + o * M_CTR + mg] = r;
            }
        }
    }
}

// ---------------------------------------------------------------------------
// Kernel 4: dense branch, fully fused per (b, m):
//   gather (layout-reinterpreted) -> pair-packed X[256 x 64] in LDS
//   WMMA GEMM 128x64, K=256  -> BN+ReLU -> max over k -> Y[128 x 32] in LDS
//   build padded im2col panel Z[384 x 16] (reusing X's LDS) once
//   1x3 stride-2 conv as WMMA GEMM 128x16, K=384 -> BN+ReLU -> dense_out
// Intermediate y (67 MB in f32) never leaves LDS.
// ---------------------------------------------------------------------------
extern "C" __global__ __launch_bounds__(256)
void dense_kernel(const float* __restrict__ df, const int* __restrict__ fpsIdx,
                  const int* __restrict__ knnIdx,
                  const float* __restrict__ Wdn, const float* __restrict__ bdn,
                  const float* __restrict__ sdn, const float* __restrict__ hdn,
                  const float* __restrict__ Wds, const float* __restrict__ bds,
                  const float* __restrict__ sds, const float* __restrict__ hds,
                  float* __restrict__ out1) {
    extern __shared__ float smem[];
    float* X = smem;                          // [128 pair-rows][XP_STRIDE f2] (reused as Z)
    float* Y = smem + 128 * XP_STRIDE * 2;    // [128][33]

    const int b = blockIdx.x >> 9;
    const int m = blockIdx.x & 511;
    const int t = threadIdx.x, lane = t & 31, wid = t >> 5;
    const int half = lane >> 4, row = lane & 15;

    const int ctr = fpsIdx[b * M_CTR + m];
    const int nb0 = knnIdx[((size_t)b * M_CTR + m) * 2 + 0];
    const int nb1 = knnIdx[((size_t)b * M_CTR + m) * 2 + 1];
    const float* db = df + (size_t)b * DN_C * N_STK * N_PNT;

    // Gather with the (p' = 2p + hi) layout reinterpretation from the reshape.
    for (int e = t; e < 256 * 64; e += 256) {
        int c = e >> 6, col = e & 63;
        int p = col >> 1, k = col & 1;
        int hi = c >> 7, c7 = c & 127;
        const float* chan = db + c7 * (N_STK * N_PNT);
        float v;
        if (p < 16) {
            int pp = 2 * p + hi;
            int nb = k ? nb1 : nb0;
            v = chan[nb * N_PNT + pp] - chan[ctr * N_PNT + pp];
        } else {
            int pp = 2 * (p - 16) + hi;
            v = chan[ctr * N_PNT + pp];
        }
        X[((c >> 1) * XP_STRIDE + col) * 2 + (c & 1)] = v;
    }
    __syncthreads();

    // Main GEMM: [128 out] x [256 c] x [64 cols=(p,k)]
    {
        const int o0 = wid * 16;
        v8f acc[4] = {};
        const v2f* Wrow = (const v2f*)(Wdn + (o0 + row) * 256);
        const v2f* Xp   = (const v2f*)X;
        for (int kk = 0; kk < 64; ++kk) {
            int ch = kk * 2 + half;
            v2f a = Wrow[ch];
#pragma unroll
            for (int ct = 0; ct < 4; ++ct) {
                v2f bf = Xp[ch * XP_STRIDE + ct * 16 + row];
                acc[ct] = wmma4(a, bf, acc[ct]);
            }
        }
#pragma unroll
        for (int ct = 0; ct < 4; ++ct) {
#pragma unroll
            for (int v = 0; v < 8; ++v) {
                int o = o0 + v + 8 * half;
                int colg = ct * 16 + row;
                float tv = acc[ct][v] + bdn[o];
                tv = fmaxf(tv * sdn[o] + hdn[o], 0.f);
                float ov = __shfl_xor(tv, 1, 32);
                float r  = fmaxf(tv, ov);
                if ((lane & 1) == 0) Y[o * 33 + (colg >> 1)] = r;   // Y[o][p]
            }
        }
    }
    __syncthreads();

    // Build padded im2col panel Z[ii=(i,r)][w] = Ypad[i][2w + r - 1],
    // pair-packed over ii, reusing X's LDS space (X is dead now).
    float* Z = X;
    for (int e = t; e < 384 * 16; e += 256) {
        int ii = e >> 4, w = e & 15;
        int i = ii / 3, r3 = ii - i * 3;
        int pw = 2 * w + r3 - 1;
        float val = (pw >= 0 && pw < 32) ? Y[i * 33 + pw] : 0.f;
        Z[((ii >> 1) * ZP_STRIDE + w) * 2 + (ii & 1)] = val;
    }
    __syncthreads();

    // Conv as GEMM: out[o][w] = sum_ii Wds[o][ii] * Z[ii][w]
    {
        const int o0 = wid * 16;
        v8f acc = {};
        const v2f* Wrow = (const v2f*)(Wds + (o0 + row) * 384);
        const v2f* Zp   = (const v2f*)Z;
        for (int kk = 0; kk < 96; ++kk) {
            int ch = kk * 2 + half;
            v2f a  = Wrow[ch];
            v2f bf = Zp[ch * ZP_STRIDE + row];
            acc = wmma4(a, bf, acc);
        }
#pragma unroll
        for (int v = 0; v < 8; ++v) {
            int o = o0 + v + 8 * half;
            float tv = acc[v] + bds[o];
            tv = fmaxf(tv * sds[o] + hds[o], 0.f);
            out1[(((size_t)b * DN_C + o) * M_CTR + m) * 16 + row] = tv;
        }
    }
}

// ---------------------------------------------------------------------------
extern "C" void kernel_launch(void* const* d_in, const int* in_sizes, int n_in,
                              void* d_out, int out_size, void* d_ws, size_t ws_size,
                              hipStream_t stream) {
    (void)in_sizes; (void)n_in; (void)out_size; (void)ws_size;
    const float* sparse_fea = (const float*)d_in[0];
    const float* dense_fea  = (const float*)d_in[1];
    const float* stk_coor   = (const float*)d_in[2];
    const float* W_sp = (const float*)d_in[3];
    const float* b_sp = (const float*)d_in[4];
    const float* s_sp = (const float*)d_in[5];
    const float* h_sp = (const float*)d_in[6];
    const float* W_dn = (const float*)d_in[7];
    const float* b_dn = (const float*)d_in[8];
    const float* s_dn = (const float*)d_in[9];
    const float* h_dn = (const float*)d_in[10];
    const float* W_ds = (const float*)d_in[11];
    const float* b_ds = (const float*)d_in[12];
    const float* s_ds = (const float*)d_in[13];
    const float* h_ds = (const float*)d_in[14];

    float* out0 = (float*)d_out;                                  // sparse_out [8,128,512]
    float* out1 = out0 + (size_t)BS * SP_C * M_CTR;               // dense_out  [8,128,512,16]
    float* out2 = out1 + (size_t)BS * DN_C * M_CTR * 16;          // coords     [8,512,32]

    int* fpsIdx = (int*)d_ws;                                     // [8,512]
    int* knnIdx = fpsIdx + BS * M_CTR;                            // [8,512,2]

    size_t sh_fps = (size_t)(N_STK * 33 + M_CTR + 32 + 32 + 4) * 4;
    fps_kernel<<<BS, 1024, sh_fps, stream>>>(stk_coor, fpsIdx, out2);

    size_t sh_knn = (size_t)(N_STK * 33 + N_STK) * 4;
    knn_kernel<<<BS * 64, 256, sh_knn, stream>>>(stk_coor, fpsIdx, knnIdx);

    size_t sh_sp = (size_t)(128 * XP_STRIDE * 2) * 4 + 96 * 4;
    sparse_kernel<<<BS * 16, 256, sh_sp, stream>>>(sparse_fea, fpsIdx, knnIdx,
                                                   W_sp, b_sp, s_sp, h_sp, out0);

    size_t sh_dn = (size_t)(128 * XP_STRIDE * 2 + 128 * 33) * 4;
    dense_kernel<<<BS * 512, 256, sh_dn, stream>>>(dense_fea, fpsIdx, knnIdx,
                                                   W_dn, b_dn, s_dn, h_dn,
                                                   W_ds, b_ds, s_ds, h_ds, out1);
}